// AtomEmbeddingAndSumLastLayer_50525995270405
// MI455X (gfx1250) — compile-verified
//
#include <hip/hip_runtime.h>

typedef __attribute__((ext_vector_type(2))) float v2f;
typedef __attribute__((ext_vector_type(8))) float v8f;
typedef __attribute__((ext_vector_type(4))) unsigned int v4u;
typedef __attribute__((ext_vector_type(4))) int v4i;
typedef __attribute__((ext_vector_type(8))) int v8i;

#define DIN  118   // one-hot dim (K)
#define DOUT 64    // embedded dim
#define ROWS 64    // atoms per block
#define TPB  128   // 4 waves (wave32)
#define HS   68    // LDS stride for h tile

__global__ void zero_kernel(float* __restrict__ p, int n) {
    int i = blockIdx.x * blockDim.x + threadIdx.x;
    if (i < n) p[i] = 0.0f;
}

// Issue one TDM 2-D tile load (global -> LDS). Wave-level op: caller must
// ensure only one wave reaches this. D# packing per CDNA5 ISA ch.8.
__device__ __forceinline__ void tdm_load_2d(unsigned lds_off_bytes,
                                            const void* gptr,
                                            unsigned tensor_d0, unsigned tensor_d1,
                                            unsigned tile_d0, unsigned tile_d1,
                                            unsigned long long stride0_elems) {
    unsigned long long ga = (unsigned long long)(size_t)gptr;
    v4u g0;
    g0[0] = 1u;                                                // count=1, user mode
    g0[1] = lds_off_bytes;                                     // lds_addr (bytes)
    g0[2] = (unsigned)(ga & 0xFFFFFFFFull);                    // global_addr[31:0]
    g0[3] = (unsigned)((ga >> 32) & 0x1FFFFFFull) | (2u << 30);// addr[56:32] | type=2

    v8i g1;
    g1[0] = (int)(2u << 16);                                   // data_size = 4 bytes
    g1[1] = (int)((tensor_d0 & 0xFFFFu) << 16);                // tensor_dim0[15:0] @48
    g1[2] = (int)(((tensor_d0 >> 16) & 0xFFFFu) |              // tensor_dim0[31:16]
                  ((tensor_d1 & 0xFFFFu) << 16));              // tensor_dim1[15:0]
    g1[3] = (int)(((tensor_d1 >> 16) & 0xFFFFu) |              // tensor_dim1[31:16]
                  ((tile_d0 & 0xFFFFu) << 16));                // tile_dim0
    g1[4] = (int)(tile_d1 & 0xFFFFu);                          // tile_dim1 (tile_dim2=0)
    g1[5] = (int)(stride0_elems & 0xFFFFFFFFull);              // tensor_dim0_stride lo
    g1[6] = (int)((stride0_elems >> 32) & 0xFFFFull);          // stride0 hi16 (stride1=0)
    g1[7] = 0;

    v4i g2 = {0, 0, 0, 0};                                     // 2-D tensor: unused
    v4i g3 = {0, 0, 0, 0};
    v8i gx = {0, 0, 0, 0, 0, 0, 0, 0};                         // extension payload
    __builtin_amdgcn_tensor_load_to_lds(g0, g1, g2, g3, gx, 0);
}

__global__ __launch_bounds__(TPB)
void gemm_scatter(const float* __restrict__ x, const float* __restrict__ W,
                  const float* __restrict__ bias, const int* __restrict__ batch,
                  float* __restrict__ acc, int N) {
    __shared__ float ldsX[ROWS * DIN];   // 30208 B, natural stride 118, reused as h
    __shared__ float ldsW[DOUT * DIN];   // 30208 B
    __shared__ int   ldsSeg[ROWS];

    const int tid = threadIdx.x;
    const int rb  = blockIdx.x * ROWS;

    if (tid < ROWS) {
        int gr = rb + tid;
        ldsSeg[tid] = (gr < N) ? batch[gr] : -1;
    }

    // Wave 0 DMAs both tiles via the Tensor Data Mover; rows past N read as
    // zero via tensor_dim1 OOB (handles the ragged final block).
    if (tid < 32) {
        tdm_load_2d((unsigned)(size_t)ldsX, x + (size_t)rb * DIN,
                    DIN, (unsigned)(N - rb), DIN, ROWS, DIN);
        tdm_load_2d((unsigned)(size_t)ldsW, W,
                    DIN, DOUT, DIN, DOUT, DIN);
        __builtin_amdgcn_s_wait_tensorcnt(0);
    }
    __syncthreads();

    const int wave = tid >> 5;
    const int lane = tid & 31;
    const int half = lane >> 4;   // K sub-pair select per ISA A/B layout
    const int lm   = lane & 15;   // M (for A) / N (for B)
    const int m0   = wave * 16;   // this wave's 16-row tile

    v8f c0 = {}, c1 = {}, c2 = {}, c3 = {};

    const float* xrow = &ldsX[(m0 + lm) * DIN + 2 * half];
    const float* w0   = &ldsW[( 0 + lm) * DIN + 2 * half];
    const float* w1   = &ldsW[(16 + lm) * DIN + 2 * half];
    const float* w2   = &ldsW[(32 + lm) * DIN + 2 * half];
    const float* w3   = &ldsW[(48 + lm) * DIN + 2 * half];

    // 29 full K-steps of 4 (K=0..115), 4 column tiles each.
    for (int k0 = 0; k0 < 116; k0 += 4) {
        v2f a  = *(const v2f*)(xrow + k0);
        v2f b0 = *(const v2f*)(w0 + k0);
        v2f b1 = *(const v2f*)(w1 + k0);
        v2f b2 = *(const v2f*)(w2 + k0);
        v2f b3 = *(const v2f*)(w3 + k0);
        c0 = __builtin_amdgcn_wmma_f32_16x16x4_f32(false, a, false, b0, (short)0, c0, false, false);
        c1 = __builtin_amdgcn_wmma_f32_16x16x4_f32(false, a, false, b1, (short)0, c1, false, false);
        c2 = __builtin_amdgcn_wmma_f32_16x16x4_f32(false, a, false, b2, (short)0, c2, false, false);
        c3 = __builtin_amdgcn_wmma_f32_16x16x4_f32(false, a, false, b3, (short)0, c3, false, false);
    }
    // K tail (116,117): lanes in the upper half would read K=118,119 -> zero.
    {
        v2f a  = {0.0f, 0.0f};
        v2f b0 = a, b1 = a, b2 = a, b3 = a;
        if (half == 0) {
            a  = *(const v2f*)(xrow + 116);
            b0 = *(const v2f*)(w0 + 116);
            b1 = *(const v2f*)(w1 + 116);
            b2 = *(const v2f*)(w2 + 116);
            b3 = *(const v2f*)(w3 + 116);
        }
        c0 = __builtin_amdgcn_wmma_f32_16x16x4_f32(false, a, false, b0, (short)0, c0, false, false);
        c1 = __builtin_amdgcn_wmma_f32_16x16x4_f32(false, a, false, b1, (short)0, c1, false, false);
        c2 = __builtin_amdgcn_wmma_f32_16x16x4_f32(false, a, false, b2, (short)0, c2, false, false);
        c3 = __builtin_amdgcn_wmma_f32_16x16x4_f32(false, a, false, b3, (short)0, c3, false, false);
    }

    // Bias + ReLU (column index per lane = tile*16 + lm).
    float bv0 = bias[ 0 + lm];
    float bv1 = bias[16 + lm];
    float bv2 = bias[32 + lm];
    float bv3 = bias[48 + lm];
#pragma unroll
    for (int v = 0; v < 8; ++v) {
        c0[v] = fmaxf(c0[v] + bv0, 0.0f);
        c1[v] = fmaxf(c1[v] + bv1, 0.0f);
        c2[v] = fmaxf(c2[v] + bv2, 0.0f);
        c3[v] = fmaxf(c3[v] + bv3, 0.0f);
    }

    __syncthreads();                 // done reading ldsX; reuse it for h
    float* ldsH = ldsX;              // [ROWS x DOUT], stride HS
#pragma unroll
    for (int v = 0; v < 8; ++v) {
        int r = m0 + v + 8 * half;   // C layout: VGPR v -> M = v + 8*half
        ldsH[r * HS +  0 + lm] = c0[v];
        ldsH[r * HS + 16 + lm] = c1[v];
        ldsH[r * HS + 32 + lm] = c2[v];
        ldsH[r * HS + 48 + lm] = c3[v];
    }
    __syncthreads();

    // Segmented (run-flush) reduction: batch is sorted, so consecutive rows
    // mostly share a segment -> ~1 atomic per (run x column) instead of per row.
    int col = tid & 63;
    int r0  = (tid >> 6) * 32;       // two 32-row slices per column
    float accv = 0.0f;
    int cur = -1;
    for (int i = 0; i < 32; ++i) {
        int r  = r0 + i;
        int gr = rb + r;
        if (gr >= N) break;
        int seg = ldsSeg[r];
        if (seg != cur) {
            if (cur >= 0) atomicAdd(&acc[(long long)cur * DOUT + col], accv);
            cur = seg;
            accv = 0.0f;
        }
        accv += ldsH[r * HS + col];
    }
    if (cur >= 0) atomicAdd(&acc[(long long)cur * DOUT + col], accv);
}

__global__ __launch_bounds__(64)
void finalize_kernel(float* __restrict__ s, int G) {
    __shared__ float red[64];
    int g = blockIdx.x;
    int t = threadIdx.x;
    if (g >= G) return;
    float v = fmaxf(s[(long long)g * DOUT + t], 0.0f);   // relu(s)
    red[t] = v;
    __syncthreads();
#pragma unroll
    for (int off = 32; off > 0; off >>= 1) {
        if (t < off) red[t] = fmaxf(red[t], red[t + off]);
        __syncthreads();
    }
    s[(long long)g * DOUT + t] = v / red[0];             // s / rowmax
}

extern "C" void kernel_launch(void* const* d_in, const int* in_sizes, int n_in,
                              void* d_out, int out_size, void* d_ws, size_t ws_size,
                              hipStream_t stream) {
    const float* x     = (const float*)d_in[0];
    const float* W     = (const float*)d_in[1];
    const float* b     = (const float*)d_in[2];
    const int*   batch = (const int*)d_in[3];
    float* out = (float*)d_out;

    const int N = in_sizes[3];           // atoms
    const int G = out_size / DOUT;       // segments

    zero_kernel<<<(out_size + 255) / 256, 256, 0, stream>>>(out, out_size);

    int blocks = (N + ROWS - 1) / ROWS;
    gemm_scatter<<<blocks, TPB, 0, stream>>>(x, W, b, batch, out, N);

    finalize_kernel<<<G, 64, 0, stream>>>(out, G);
}